// TopKGate_72464688218816
// MI455X (gfx1250) — compile-verified
//
#include <hip/hip_runtime.h>
#include <hip/hip_bf16.h>

typedef float v2f __attribute__((ext_vector_type(2)));
typedef float v4f __attribute__((ext_vector_type(4)));
typedef float v8f __attribute__((ext_vector_type(8)));

#define T_TOK 4096
#define H_DIM 4096
#define E_EXP 8
#define K_TOP 2
#define CAP   1280                       // int(4096*2/8*1.25)
#define TEC   ((size_t)T_TOK*E_EXP*CAP)  // 41,943,040 per [T,E,C] tensor

// -------------------------------------------------------------------------
// Kernel Z: stream 335MB of zeros (dispatch_mask + combine_weights) with
// non-temporal stores; also zero the 16 scalar accumulators in workspace.
// -------------------------------------------------------------------------
__global__ __launch_bounds__(256)
void zero_fill(float* __restrict__ out, float* __restrict__ scal) {
    const size_t n4 = (2 * TEC) / 4;
    size_t i = (size_t)blockIdx.x * 256 + threadIdx.x;
    size_t stride = (size_t)gridDim.x * 256;
    v4f z = {0.f, 0.f, 0.f, 0.f};
    v4f* o = (v4f*)out;
    for (size_t j = i; j < n4; j += stride)
        __builtin_nontemporal_store(z, o + j);
    if (blockIdx.x == 0 && threadIdx.x < 16) scal[threadIdx.x] = 0.f;
}

// -------------------------------------------------------------------------
// Kernel A: fused gate GEMM (V_WMMA_F32_16X16X4_F32) + per-token top-2 +
// softmax stats. 8 waves/block, each wave owns a 16-token tile.
//   A = W tile  : 16(experts, rows 8..15 = 0) x 4(K)   [from LDS]
//   B = x tile  : 4(K) x 16(tokens)                    [from global, NT]
//   D[M=e][N=tok]: lane n (<16) holds all 8 expert logits in acc[0..7].
// Deep unroll (16) keeps 16 global b64 loads in flight per wave so the
// 64MB streaming read of x stays bandwidth-bound, not latency-bound.
// -------------------------------------------------------------------------
__global__ __launch_bounds__(256)
void gate_gemm_topk(const float* __restrict__ x, const float* __restrict__ W,
                    float* __restrict__ out_topi,
                    float* __restrict__ sumProbs, float* __restrict__ zsum,
                    int* __restrict__ eIdx, float* __restrict__ wVal) {
    extern __shared__ float lds[];  // [8*H_DIM] W + [4] zeros + [8] psum + [1] zls
    const int tid = threadIdx.x;

    // Stage W (8x4096 f32 = 128KB) into LDS: 8192 float4's over 256 threads.
    {
        const v4f* Wv = (const v4f*)W;
        v4f* Lv = (v4f*)lds;
#pragma unroll
        for (int i = 0; i < 32; ++i) Lv[tid + i * 256] = Wv[tid + i * 256];
    }
    if (tid < 16) lds[8 * H_DIM + tid] = 0.0f;  // zero pad + loss accumulators
    __syncthreads();

    const int lane = tid & 31;
    const int wave = tid >> 5;
    const int m    = lane & 15;   // A: expert row / B: token column
    const int l16  = lane >> 4;   // K half select
    const int tokBase = blockIdx.x * 128 + wave * 16;

    // A-operand (W) LDS float offsets; rows >= 8 read the constant zero pad.
    int aOff = (m < 8) ? (m * H_DIM + 2 * l16) : (8 * H_DIM);
    const int aStep = (m < 8) ? 4 : 0;
    const float* bPtr = x + (size_t)(tokBase + m) * H_DIM + 2 * l16;

    v8f acc = {};
#pragma unroll 16
    for (int k0 = 0; k0 < H_DIM; k0 += 4) {
        v2f a = *(const v2f*)(lds + aOff);                       // ds_load_b64
        v2f b = __builtin_nontemporal_load((const v2f*)(bPtr + k0));  // NT global_load_b64
        acc = __builtin_amdgcn_wmma_f32_16x16x4_f32(
            false, a, false, b, (short)0, acc, false, false);
        aOff += aStep;
    }

    // Lane-local epilogue: lane n (<16) owns token tokBase+n, logits acc[0..7].
    if (l16 == 0) {
        float l[8];
#pragma unroll
        for (int e = 0; e < 8; ++e) l[e] = acc[e];

        int i0 = 0; float v0 = l[0];
#pragma unroll
        for (int e = 1; e < 8; ++e) if (l[e] > v0) { v0 = l[e]; i0 = e; }
        int i1 = -1; float v1 = -3.402823466e38f;
#pragma unroll
        for (int e = 0; e < 8; ++e) if (e != i0 && l[e] > v1) { v1 = l[e]; i1 = e; }

        float s  = __expf(v1 - v0);            // softmax over top-2
        float w0 = 1.0f / (1.0f + s);
        float w1 = 1.0f - w0;

        float sum = 0.f, p[8];
#pragma unroll
        for (int e = 0; e < 8; ++e) { p[e] = __expf(l[e] - v0); sum += p[e]; }
        float inv = 1.0f / sum;
        float lse = v0 + __logf(sum);

        const int t = tokBase + m;
        out_topi[t * 2]     = (float)i0;
        out_topi[t * 2 + 1] = (float)i1;
        eIdx[t * 2] = i0;  eIdx[t * 2 + 1] = i1;
        wVal[t * 2] = w0;  wVal[t * 2 + 1] = w1;

#pragma unroll
        for (int e = 0; e < 8; ++e) atomicAdd(&lds[8 * H_DIM + 4 + e], p[e] * inv);
        atomicAdd(&lds[8 * H_DIM + 12], lse);
    }
    __syncthreads();
    if (tid < 8)       atomicAdd(&sumProbs[tid], lds[8 * H_DIM + 4 + tid]);
    else if (tid == 8) atomicAdd(zsum, lds[8 * H_DIM + 12]);
}

// -------------------------------------------------------------------------
// Kernel B: single-workgroup ordered exclusive scan over the 8192 (token,k)
// assignments (8-wide per-expert counts, Hillis-Steele over 1024 threads),
// producing deterministic capacity slots; thread 0 finalizes both losses.
// -------------------------------------------------------------------------
__global__ __launch_bounds__(1024)
void scan_assign(const int* __restrict__ eIdx, const float* __restrict__ wVal,
                 const float* __restrict__ sumProbs, const float* __restrict__ zsum,
                 int* __restrict__ sIdx, float* __restrict__ sW, int* __restrict__ sV,
                 float* __restrict__ out_losses) {
    __shared__ int sc[1024 * 8];
    const int tid = threadIdx.x;
    const int base = tid * 8;  // 8 assignments per thread, in flat order

    int cnt[8] = {0, 0, 0, 0, 0, 0, 0, 0};
    int es[8];
#pragma unroll
    for (int j = 0; j < 8; ++j) { int e = eIdx[base + j]; es[j] = e; cnt[e]++; }
#pragma unroll
    for (int e = 0; e < 8; ++e) sc[tid * 8 + e] = cnt[e];
    __syncthreads();

    for (int off = 1; off < 1024; off <<= 1) {
        int v[8];
        if (tid >= off) {
#pragma unroll
            for (int e = 0; e < 8; ++e) v[e] = sc[(tid - off) * 8 + e];
        }
        __syncthreads();
        if (tid >= off) {
#pragma unroll
            for (int e = 0; e < 8; ++e) sc[tid * 8 + e] += v[e];
        }
        __syncthreads();
    }

    int run[8];
#pragma unroll
    for (int e = 0; e < 8; ++e) run[e] = (tid > 0) ? sc[(tid - 1) * 8 + e] : 0;

#pragma unroll
    for (int j = 0; j < 8; ++j) {
        int e = es[j];
        int pos = run[e]++;
        int valid = pos < CAP;
        int posc = valid ? pos : (CAP - 1);
        int tok = (base + j) >> 1;  // flat index / K
        sIdx[base + j] = tok * (E_EXP * CAP) + e * CAP + posc;
        sW[base + j]   = valid ? wVal[base + j] : 0.0f;
        sV[base + j]   = valid;
    }

    if (tid == 0) {
        float tpe[8], tsum = 0.f;
#pragma unroll
        for (int e = 0; e < 8; ++e) {
            int tot = sc[1023 * 8 + e];
            tpe[e] = (float)(tot < CAP ? tot : CAP);
            tsum += tpe[e];
        }
        float lb = 0.f;
#pragma unroll
        for (int e = 0; e < 8; ++e)
            lb += (sumProbs[e] / (float)T_TOK) * (tpe[e] / tsum);
        out_losses[0] = 0.01f * (float)E_EXP * lb;          // load balancing loss
        out_losses[1] = 0.001f * (zsum[0] / (float)T_TOK);  // router z loss
    }
}

// -------------------------------------------------------------------------
// Kernel S: scatter the 8192 nonzeros (flat indices are unique: one per
// (token, expert) pair, since top-k indices are distinct).
// -------------------------------------------------------------------------
__global__ __launch_bounds__(256)
void scatter_out(const int* __restrict__ sIdx, const float* __restrict__ sW,
                 const int* __restrict__ sV,
                 float* __restrict__ dispatch, float* __restrict__ combine) {
    int j = blockIdx.x * 256 + threadIdx.x;
    if (j < T_TOK * K_TOP) {
        int idx = sIdx[j];
        combine[idx]  = sW[j];
        dispatch[idx] = sV[j] ? 1.0f : 0.0f;
    }
}

// -------------------------------------------------------------------------
extern "C" void kernel_launch(void* const* d_in, const int* in_sizes, int n_in,
                              void* d_out, int out_size, void* d_ws, size_t ws_size,
                              hipStream_t stream) {
    const float* x = (const float*)d_in[0];   // [4,1024,4096] f32
    const float* W = (const float*)d_in[1];   // [8,4096] f32

    float* out      = (float*)d_out;
    float* dispatch = out;                          // [T,E,C]
    float* combine  = out + TEC;                    // [T,E,C]
    float* topi     = out + 2 * TEC;                // [T,K]
    float* losses   = topi + (size_t)T_TOK * K_TOP; // 2 scalars

    // workspace layout (floats): [0..7] sumProbs, [8] zsum, [16..] arrays
    float* p        = (float*)d_ws;
    float* sumProbs = p;
    float* zsum     = p + 8;
    int*   eIdx     = (int*)(p + 16);
    float* wVal     = p + 16 + 8192;
    int*   sIdx     = (int*)(p + 16 + 2 * 8192);
    float* sW       = p + 16 + 3 * 8192;
    int*   sV       = (int*)(p + 16 + 4 * 8192);

    // 1) zero dispatch+combine (335MB, NT stores) + workspace accumulators
    zero_fill<<<2048, 256, 0, stream>>>(out, p);
    // 2) fused GEMM (f32 WMMA) + top-2 + softmax stats; 128KB dynamic LDS
    gate_gemm_topk<<<32, 256, (8 * H_DIM + 16) * sizeof(float), stream>>>(
        x, W, topi, sumProbs, zsum, eIdx, wVal);
    // 3) deterministic ordered scan -> capacity slots + losses
    scan_assign<<<1, 1024, 0, stream>>>(eIdx, wVal, sumProbs, zsum,
                                        sIdx, sW, sV, losses);
    // 4) scatter the 8192 nonzeros
    scatter_out<<<32, 256, 0, stream>>>(sIdx, sW, sV, dispatch, combine);
}